// MicroEncoder_23476291240585
// MI455X (gfx1250) — compile-verified
//
#include <hip/hip_runtime.h>

// ---------------------------------------------------------------------------
// DCRNN (2-layer diffusion-conv GRU, H0 = 0) for MI455X / gfx1250.
// Key simplifications (exact, from H0 == 0 in the reference):
//   * R gate never affects the output (R*H == 0)  -> Wr/br unused.
//   * Only the X-half (rows 0..31) of each 64x32 weight block is live.
//   * cell(X) = (1 - sigmoid(Gz)) * tanh(Gh),
//     Gg = X@(Wg[0,0]+Wg[1,0])[:32] + Pout(X)@Wg[0,1][:32]
//        + Pin(X)@Wg[1,1][:32] + bg
// Dense GEMMs run on V_WMMA_F32_16X16X4_F32 (exact fp32).
// ---------------------------------------------------------------------------

typedef __attribute__((ext_vector_type(2))) float v2f;
typedef __attribute__((ext_vector_type(8))) float v8f;

#define NNODES 100000
#define NEDGES 1600000
#define NTILES 6250          // NNODES / 16 exactly

// ---------------------------- degree accumulation ---------------------------
__global__ void deg_kernel(const int* __restrict__ src, const int* __restrict__ dst,
                           const float* __restrict__ w,
                           float* __restrict__ deg_out, float* __restrict__ deg_in,
                           int E) {
  int e = blockIdx.x * blockDim.x + threadIdx.x;
  if (e < E) {
    float we = w[e];
    atomicAdd(deg_out + src[e], we);
    atomicAdd(deg_in  + dst[e], we);
  }
}

__global__ void norm_kernel(const int* __restrict__ src, const int* __restrict__ dst,
                            const float* __restrict__ w,
                            const float* __restrict__ deg_out, const float* __restrict__ deg_in,
                            float* __restrict__ norm_out, float* __restrict__ norm_in,
                            int E) {
  int e = blockIdx.x * blockDim.x + threadIdx.x;
  if (e < E) {
    float we = w[e];
    norm_out[e] = we / deg_out[src[e]];
    norm_in[e]  = we / deg_in[dst[e]];
  }
}

// ------------------------- sparse diffusion hop ----------------------------
// Out[sidx[e]] += norm[e] * X[gidx[e]]   (32 channels, 8 threads per edge)
__global__ void prop_kernel(const float* __restrict__ X, const float* __restrict__ norm,
                            const int* __restrict__ gidx, const int* __restrict__ sidx,
                            float* __restrict__ Out, int E) {
  int gid = blockIdx.x * blockDim.x + threadIdx.x;
  int e  = gid >> 3;
  int c0 = (gid & 7) << 2;
  if (e < E) {
    float n = norm[e];
    int g = gidx[e];
    int s = sidx[e];
    float4 xv = *reinterpret_cast<const float4*>(X + (size_t)g * 32 + c0);
    float* op = Out + (size_t)s * 32 + c0;
    atomicAdd(op + 0, n * xv.x);
    atomicAdd(op + 1, n * xv.y);
    atomicAdd(op + 2, n * xv.z);
    atomicAdd(op + 3, n * xv.w);
  }
}

// --------------------- fused WMMA gate kernel (Z & H) ----------------------
// One wave computes a 16-node x 32-col output tile:
//   G = [X | Sout | Sin] (16x96)  @  Wc (96x32)      for both gates
//   out = (1 - sigmoid(Gz + bz)) * tanh(Gh + bh)     (optional ReLU)
__global__ __launch_bounds__(256)
void gate_kernel(const float* __restrict__ X, const float* __restrict__ Sout,
                 const float* __restrict__ Sin,
                 const float* __restrict__ Wz, const float* __restrict__ bz,
                 const float* __restrict__ Wh, const float* __restrict__ bh,
                 float* __restrict__ Out, int apply_relu) {
  __shared__ float WzC[96 * 32];   // rows 0..31: W[0,0]+W[1,0]; 32..63: W[0,1]; 64..95: W[1,1]
  __shared__ float WhC[96 * 32];

  // W flat layout: d*4096 + k*2048 + r*32 + c   (shape (2,2,64,32))
  for (int i = threadIdx.x; i < 96 * 32; i += blockDim.x) {
    int r = i >> 5, c = i & 31;
    float vz, vh;
    if (r < 32) {
      vz = Wz[r * 32 + c] + Wz[4096 + r * 32 + c];
      vh = Wh[r * 32 + c] + Wh[4096 + r * 32 + c];
    } else if (r < 64) {
      int rr = r - 32;
      vz = Wz[2048 + rr * 32 + c];
      vh = Wh[2048 + rr * 32 + c];
    } else {
      int rr = r - 64;
      vz = Wz[6144 + rr * 32 + c];
      vh = Wh[6144 + rr * 32 + c];
    }
    WzC[i] = vz;
    WhC[i] = vh;
  }
  __syncthreads();

  int wave = threadIdx.x >> 5;
  int lane = threadIdx.x & 31;
  int tile = blockIdx.x * 8 + wave;
  if (tile >= NTILES) return;          // wave-uniform: EXEC all-ones below

  int half = lane >> 4;                // K/M sub-select per ISA layout
  int l16  = lane & 15;
  int m0   = tile * 16;

  v8f accz0 = {}, accz1 = {}, acch0 = {}, acch1 = {};

  const float* srcs[3] = {X, Sout, Sin};
  const float* arow_base = nullptr;
  for (int blk = 0; blk < 3; ++blk) {
    arow_base = srcs[blk] + (size_t)(m0 + l16) * 32;   // A row: M = lane%16
#pragma unroll
    for (int k4 = 0; k4 < 8; ++k4) {
      int ca = k4 * 4 + half * 2;                      // K within 32-col block
      int kk = blk * 32 + k4 * 4;                      // K into 96-row Wc
      v2f a;
      a.x = arow_base[ca];
      a.y = arow_base[ca + 1];

      const float* wz0 = WzC + (kk + half * 2) * 32;
      const float* wz1 = wz0 + 32;
      const float* wh0 = WhC + (kk + half * 2) * 32;
      const float* wh1 = wh0 + 32;
      v2f bza, bzb, bha, bhb;
      bza.x = wz0[l16];      bza.y = wz1[l16];         // N-tile 0 (cols 0..15)
      bzb.x = wz0[16 + l16]; bzb.y = wz1[16 + l16];    // N-tile 1 (cols 16..31)
      bha.x = wh0[l16];      bha.y = wh1[l16];
      bhb.x = wh0[16 + l16]; bhb.y = wh1[16 + l16];

      accz0 = __builtin_amdgcn_wmma_f32_16x16x4_f32(false, a, false, bza, (short)0, accz0, false, false);
      accz1 = __builtin_amdgcn_wmma_f32_16x16x4_f32(false, a, false, bzb, (short)0, accz1, false, false);
      acch0 = __builtin_amdgcn_wmma_f32_16x16x4_f32(false, a, false, bha, (short)0, acch0, false, false);
      acch1 = __builtin_amdgcn_wmma_f32_16x16x4_f32(false, a, false, bhb, (short)0, acch1, false, false);
    }
  }

  float bz0 = bz[l16], bz1 = bz[16 + l16];
  float bh0 = bh[l16], bh1 = bh[16 + l16];
#pragma unroll
  for (int v = 0; v < 8; ++v) {
    int node = m0 + v + half * 8;                      // D: M = v + 8*(lane/16)
    float z  = 1.0f / (1.0f + expf(-(accz0[v] + bz0)));
    float ht = tanhf(acch0[v] + bh0);
    float o  = (1.0f - z) * ht;
    if (apply_relu) o = fmaxf(o, 0.0f);
    Out[(size_t)node * 32 + l16] = o;

    z  = 1.0f / (1.0f + expf(-(accz1[v] + bz1)));
    ht = tanhf(acch1[v] + bh1);
    o  = (1.0f - z) * ht;
    if (apply_relu) o = fmaxf(o, 0.0f);
    Out[(size_t)node * 32 + 16 + l16] = o;
  }
}

// ------------------------------- launcher ----------------------------------
extern "C" void kernel_launch(void* const* d_in, const int* in_sizes, int n_in,
                              void* d_out, int out_size, void* d_ws, size_t ws_size,
                              hipStream_t stream) {
  const float* x    = (const float*)d_in[0];
  const int*   eidx = (const int*)  d_in[1];
  const float* ew   = (const float*)d_in[2];
  const float* Wz1  = (const float*)d_in[3];
  const float* bz1  = (const float*)d_in[4];
  // d_in[5], d_in[6] = Wr1, br1 : mathematically unused (H0 == 0)
  const float* Wh1  = (const float*)d_in[7];
  const float* bh1  = (const float*)d_in[8];
  const float* Wz2  = (const float*)d_in[9];
  const float* bz2  = (const float*)d_in[10];
  // d_in[11], d_in[12] = Wr2, br2 : unused
  const float* Wh2  = (const float*)d_in[13];
  const float* bh2  = (const float*)d_in[14];

  const int* src = eidx;            // edge_index[0]
  const int* dst = eidx + NEDGES;   // edge_index[1]

  float* ws       = (float*)d_ws;
  float* deg_out  = ws;                         // [NNODES]
  float* deg_in   = deg_out + NNODES;           // [NNODES]
  float* norm_out = deg_in + NNODES;            // [NEDGES]
  float* norm_in  = norm_out + NEDGES;          // [NEDGES]
  float* Sout     = norm_in + NEDGES;           // [NNODES*32]
  float* Sin      = Sout + (size_t)NNODES * 32; // [NNODES*32]
  float* H1       = Sin  + (size_t)NNODES * 32; // [NNODES*32]
  float* out      = (float*)d_out;

  const int TB = 256;
  const int edge_blocks = (NEDGES + TB - 1) / TB;       // 6250
  const int prop_blocks = (NEDGES * 8) / TB;            // 50000 exact
  const int gate_blocks = (NTILES + 7) / 8;             // 782

  // degrees + per-edge norms (dispatch-invariant, but recomputed each call)
  hipMemsetAsync(deg_out, 0, (size_t)2 * NNODES * sizeof(float), stream);
  deg_kernel<<<edge_blocks, TB, 0, stream>>>(src, dst, ew, deg_out, deg_in, NEDGES);
  norm_kernel<<<edge_blocks, TB, 0, stream>>>(src, dst, ew, deg_out, deg_in,
                                              norm_out, norm_in, NEDGES);

  // ---- cell 1: input x -> H1 (with ReLU) ----
  hipMemsetAsync(Sout, 0, (size_t)2 * NNODES * 32 * sizeof(float), stream);
  prop_kernel<<<prop_blocks, TB, 0, stream>>>(x, norm_out, src, dst, Sout, NEDGES);
  prop_kernel<<<prop_blocks, TB, 0, stream>>>(x, norm_in,  dst, src, Sin,  NEDGES);
  gate_kernel<<<gate_blocks, TB, 0, stream>>>(x, Sout, Sin, Wz1, bz1, Wh1, bh1, H1, 1);

  // ---- cell 2: input H1 -> out (no ReLU) ----
  hipMemsetAsync(Sout, 0, (size_t)2 * NNODES * 32 * sizeof(float), stream);
  prop_kernel<<<prop_blocks, TB, 0, stream>>>(H1, norm_out, src, dst, Sout, NEDGES);
  prop_kernel<<<prop_blocks, TB, 0, stream>>>(H1, norm_in,  dst, src, Sin,  NEDGES);
  gate_kernel<<<gate_blocks, TB, 0, stream>>>(H1, Sout, Sin, Wz2, bz2, Wh2, bh2, out, 0);
}